// VisionTransformer_1228360647095
// MI455X (gfx1250) — compile-verified
//
#include <hip/hip_runtime.h>
#include <math.h>

typedef __attribute__((ext_vector_type(2))) float v2f;
typedef __attribute__((ext_vector_type(8))) float v8f;

__device__ __forceinline__ v8f wmma4(v2f a, v2f b, v8f c) {
  // V_WMMA_F32_16X16X4_F32 : D = A(16x4,f32) x B(4x16,f32) + C(16x16,f32)
  return __builtin_amdgcn_wmma_f32_16x16x4_f32(false, a, false, b, (short)0, c, false, false);
}

__device__ __forceinline__ float gelu_f(float x) {
  return 0.5f * x * (1.0f + erff(x * 0.70710678118654752f));
}

__device__ __forceinline__ v8f vzero8() {
  v8f z = {0.f,0.f,0.f,0.f,0.f,0.f,0.f,0.f};
  return z;
}

// ---- NT output tiles (16 x NT*16) per wave: C = A(MxK,LDA) * B(KxN,LDB) ----
// Compile-time strides -> strength-reduced 32-bit offsets, immediate-offset loads.
// All loads unconditional; M/N raggedness handled by index clamping (clamped
// lanes compute valid-but-unused values that are never stored). K % 4 == 0.
template<int NT, int LDA, int LDB>
__device__ __forceinline__ void mm_tiles(const float* __restrict__ A, int Mrows,
                                         const float* __restrict__ B, int Ncols,
                                         int i0, int j0, int K, v8f acc[NT]) {
  const int lane  = threadIdx.x & 31;
  const int idx16 = lane & 15;
  const int kh    = (lane >> 4) << 1;          // 0 or 2
  const int ar    = min(i0 + idx16, Mrows - 1);
  const float* Arow = A + ar * LDA + kh;
  int bc[NT];
#pragma unroll
  for (int t = 0; t < NT; ++t) bc[t] = min(j0 + t * 16 + idx16, Ncols - 1);
  int bo = kh * LDB;                           // running B offset (32-bit)
#pragma unroll 2
  for (int k = 0; k < K; k += 4) {
    v2f a = *(const v2f*)(Arow + k);           // 8B aligned: LDA even, k%4==0, kh in {0,2}
    v2f bf[NT];
#pragma unroll
    for (int t = 0; t < NT; ++t) {             // load phase: one clause, distinct regs
      bf[t].x = B[bo + bc[t]];
      bf[t].y = B[bo + LDB + bc[t]];
    }
#pragma unroll
    for (int t = 0; t < NT; ++t)               // compute phase
      acc[t] = wmma4(a, bf[t], acc[t]);
    bo += 4 * LDB;
  }
}

// Same, B given transposed: Bt is (N x K) row-major with stride LDBT. C = A * Bt^T.
template<int NT, int LDA, int LDBT>
__device__ __forceinline__ void mm_tiles_bt(const float* __restrict__ A, int Mrows,
                                            const float* __restrict__ Bt, int Ncols,
                                            int i0, int j0, int K, v8f acc[NT]) {
  const int lane  = threadIdx.x & 31;
  const int idx16 = lane & 15;
  const int kh    = (lane >> 4) << 1;
  const int ar    = min(i0 + idx16, Mrows - 1);
  const float* Arow = A + ar * LDA + kh;
  int br[NT];
#pragma unroll
  for (int t = 0; t < NT; ++t)
    br[t] = min(j0 + t * 16 + idx16, Ncols - 1) * LDBT + kh;
#pragma unroll 2
  for (int k = 0; k < K; k += 4) {
    v2f a = *(const v2f*)(Arow + k);
    v2f bf[NT];
#pragma unroll
    for (int t = 0; t < NT; ++t) bf[t] = *(const v2f*)(Bt + br[t] + k);
#pragma unroll
    for (int t = 0; t < NT; ++t) acc[t] = wmma4(a, bf[t], acc[t]);
  }
}

// ---------------- patch embedding + cls + pos ----------------
__global__ void k_patch(const float* __restrict__ x, const float* __restrict__ cw,
                        const float* __restrict__ cb, const float* __restrict__ cls,
                        const float* __restrict__ pos, float* __restrict__ act,
                        int B, int D, int S) {
  int idx = blockIdx.x * blockDim.x + threadIdx.x;
  int total = B * S * D;
  if (idx >= total) return;
  int d = idx % D;
  int s = (idx / D) % S;
  int b = idx / (D * S);
  float r;
  if (s == 0) {
    r = cls[d] + pos[d];
  } else {
    int p = s - 1, pi = p >> 3, pj = p & 7;
    float acc = cb[d];
    for (int c = 0; c < 3; ++c)
      for (int i = 0; i < 4; ++i)
        for (int j = 0; j < 4; ++j)
          acc += x[((b * 3 + c) * 32 + (pi * 4 + i)) * 32 + (pj * 4 + j)] *
                 cw[((d * 3 + c) * 4 + i) * 4 + j];
    r = acc + pos[s * D + d];
  }
  act[(size_t)(b * S + s) * D + d] = r;
}

// ---------------- LayerNorm: one 256-thread block per row (D==256) ----------------
__global__ void k_ln(const float* __restrict__ x, const float* __restrict__ g,
                     const float* __restrict__ b, float* __restrict__ out, int D) {
  int row = blockIdx.x, t = threadIdx.x;
  __shared__ float red[256];
  float v = x[(size_t)row * D + t];
  red[t] = v; __syncthreads();
  for (int s = 128; s > 0; s >>= 1) { if (t < s) red[t] += red[t + s]; __syncthreads(); }
  float mean = red[0] / (float)D;
  __syncthreads();
  float d = v - mean;
  red[t] = d * d; __syncthreads();
  for (int s = 128; s > 0; s >>= 1) { if (t < s) red[t] += red[t + s]; __syncthreads(); }
  float var = red[0] / (float)D;
  float r = rsqrtf(var + 1e-5f);
  out[(size_t)row * D + t] = d * r * g[t] + b[t];
}

// ---------------- QKV: per (b,h): Z(SxD) @ Wqkv[h](Dx3D) + bqkv[h] ----------------
__global__ void k_qkv(const float* __restrict__ Z, const float* __restrict__ W,
                      const float* __restrict__ bias, float* __restrict__ QKV,
                      int H, int S) {
  constexpr int D = 256, N = 768;
  int j0 = blockIdx.x * 64, i0 = blockIdx.y * 16, g = blockIdx.z;
  int b = g / H, h = g - b * H;
  const float* A  = Z + (size_t)b * S * D;
  const float* Wm = W + (size_t)h * D * N;
  v8f acc[4];
#pragma unroll
  for (int t = 0; t < 4; ++t) acc[t] = vzero8();
  mm_tiles<4, D, N>(A, S, Wm, N, i0, j0, D, acc);
  int lane = threadIdx.x & 31;
  int idx16 = lane & 15;
  int rbase = i0 + ((lane >> 4) << 3);
  float* Cp = QKV + (size_t)g * S * N;
  if (i0 + 16 <= S) {                      // uniform full-tile fast path
#pragma unroll
    for (int t = 0; t < 4; ++t) {
      int col = j0 + t * 16 + idx16;
      float bv = bias[h * N + col];
#pragma unroll
      for (int r = 0; r < 8; ++r)
        Cp[(size_t)(rbase + r) * N + col] = acc[t][r] + bv;
    }
  } else {
#pragma unroll
    for (int t = 0; t < 4; ++t) {
      int col = j0 + t * 16 + idx16;
      float bv = bias[h * N + col];
#pragma unroll
      for (int r = 0; r < 8; ++r) {
        int row = rbase + r;
        if (row < S) Cp[(size_t)row * N + col] = acc[t][r] + bv;
      }
    }
  }
}

// ---------------- scores: per g: Q(SxD) @ K(SxD)^T * scale, padded ld ----------------
__global__ void k_scores(const float* __restrict__ QKV, float* __restrict__ ATT,
                         int S, float scale) {
  constexpr int D = 256, N3 = 768, Spad = 68;
  int i0 = blockIdx.y * 16, g = blockIdx.z;
  const float* Q  = QKV + (size_t)g * S * N3;  // q at cols [0,D)
  const float* Kt = Q + D;                     // k at cols [D,2D): row t = K[t][:]
  v8f acc[5];
#pragma unroll
  for (int t = 0; t < 5; ++t) acc[t] = vzero8();
  mm_tiles_bt<5, N3, N3>(Q, S, Kt, S, i0, 0, D, acc);
  int lane = threadIdx.x & 31;
  int idx16 = lane & 15;
  int rbase = i0 + ((lane >> 4) << 3);
  float* Cp = ATT + (size_t)g * S * Spad;
  bool fullM = (i0 + 16 <= S);                 // uniform
#pragma unroll
  for (int t = 0; t < 5; ++t) {
    int col = t * 16 + idx16;
    // t<4: col<64 -> always in [0,S); t==4: cols 64..79, store pad zeros up to Spad
    bool cok = (t < 4) || (col < Spad);        // constant-folded per unrolled t
    float val0 = (t < 4) ? 1.f : 0.f;          // marker unused; see below
    (void)val0;
    if (cok) {
      if (fullM) {
#pragma unroll
        for (int r = 0; r < 8; ++r) {
          float v = (t < 4 || col < S) ? acc[t][r] * scale : 0.f;
          Cp[(size_t)(rbase + r) * Spad + col] = v;
        }
      } else {
#pragma unroll
        for (int r = 0; r < 8; ++r) {
          int row = rbase + r;
          float v = (t < 4 || col < S) ? acc[t][r] * scale : 0.f;
          if (row < S) Cp[(size_t)row * Spad + col] = v;
        }
      }
    }
  }
}

// ---------------- softmax: one wave32 per row of length S (stride Spad) ----------------
__global__ void k_softmax(float* __restrict__ ATT, int rows, int S, int Spad) {
  int wid = threadIdx.x >> 5;
  int lane = threadIdx.x & 31;
  int row = blockIdx.x * (blockDim.x >> 5) + wid;
  if (row >= rows) return;  // wave-uniform
  float* p = ATT + (size_t)row * Spad;
  float m = -3.4e38f;
  for (int j = lane; j < S; j += 32) m = fmaxf(m, p[j]);
  for (int off = 16; off; off >>= 1) m = fmaxf(m, __shfl_xor(m, off, 32));
  float sum = 0.f;
  for (int j = lane; j < S; j += 32) { float e = __expf(p[j] - m); p[j] = e; sum += e; }
  for (int off = 16; off; off >>= 1) sum += __shfl_xor(sum, off, 32);
  float inv = 1.f / sum;
  for (int j = lane; j < S; j += 32) p[j] *= inv;
}

// ---------------- o = attn(SxSpad) @ V(SxD), scattered to (B,S,H*D) ----------------
// K runs to Spad; pad attn columns are exactly 0 so the extra V rows contribute 0.
__global__ void k_av(const float* __restrict__ ATT, const float* __restrict__ QKV,
                     float* __restrict__ O, int H, int S) {
  constexpr int D = 256, N3 = 768, Spad = 68, HD = 4096;
  int j0 = blockIdx.x * 64, i0 = blockIdx.y * 16, g = blockIdx.z;
  int b = g / H, h = g - b * H;
  const float* A = ATT + (size_t)g * S * Spad;
  const float* V = QKV + (size_t)g * S * N3 + 2 * D;  // v at cols [2D,3D)
  v8f acc[4];
#pragma unroll
  for (int t = 0; t < 4; ++t) acc[t] = vzero8();
  mm_tiles<4, Spad, N3>(A, S, V, D, i0, j0, Spad, acc);
  int lane = threadIdx.x & 31;
  int idx16 = lane & 15;
  int rbase = i0 + ((lane >> 4) << 3);
  float* Cp = O + (size_t)b * S * HD + h * D;
  if (i0 + 16 <= S) {                      // uniform full-tile fast path
#pragma unroll
    for (int t = 0; t < 4; ++t) {
      int col = j0 + t * 16 + idx16;
#pragma unroll
      for (int r = 0; r < 8; ++r)
        Cp[(size_t)(rbase + r) * HD + col] = acc[t][r];
    }
  } else {
#pragma unroll
    for (int t = 0; t < 4; ++t) {
      int col = j0 + t * 16 + idx16;
#pragma unroll
      for (int r = 0; r < 8; ++r) {
        int row = rbase + r;
        if (row < S) Cp[(size_t)row * HD + col] = acc[t][r];
      }
    }
  }
}

// ------- generic fused GEMM (4 N-tiles/wave): C = [gelu](A@W + bias) [+ resid] -------
// ldc == N for all uses. LDA/LDB/K/N compile-time.
template<int LDA, int LDB, int K, int N, bool GELU, bool RESID>
__global__ void k_gemm4(const float* __restrict__ A, int Mrows,
                        const float* __restrict__ W,
                        const float* __restrict__ bias,
                        float* __restrict__ C,
                        const float* __restrict__ resid) {
  int j0 = blockIdx.x * 64, i0 = blockIdx.y * 16;
  v8f acc[4];
#pragma unroll
  for (int t = 0; t < 4; ++t) acc[t] = vzero8();
  mm_tiles<4, LDA, LDB>(A, Mrows, W, N, i0, j0, K, acc);
  int lane = threadIdx.x & 31;
  int idx16 = lane & 15;
  int rbase = i0 + ((lane >> 4) << 3);
  bool fullM = (i0 + 16 <= Mrows);             // uniform
#pragma unroll
  for (int t = 0; t < 4; ++t) {
    int col = j0 + t * 16 + idx16;
    bool cok = ((N & 63) == 0) || (col < N);   // constant-folds to true when N%64==0
    if (cok) {
      float bv = bias[col];
      if (fullM) {
#pragma unroll
        for (int r = 0; r < 8; ++r) {
          float v = acc[t][r] + bv;
          if (GELU) v = gelu_f(v);
          if (RESID) v += resid[(size_t)(rbase + r) * N + col];
          C[(size_t)(rbase + r) * N + col] = v;
        }
      } else {
#pragma unroll
        for (int r = 0; r < 8; ++r) {
          int row = rbase + r;
          if (row < Mrows) {
            float v = acc[t][r] + bv;
            if (GELU) v = gelu_f(v);
            if (RESID) v += resid[(size_t)row * N + col];
            C[(size_t)row * N + col] = v;
          }
        }
      }
    }
  }
}

extern "C" void kernel_launch(void* const* d_in, const int* in_sizes, int n_in,
                              void* d_out, int out_size, void* d_ws, size_t ws_size,
                              hipStream_t stream) {
  const float* x      = (const float*)d_in[0];
  const float* conv_w = (const float*)d_in[1];
  const float* conv_b = (const float*)d_in[2];
  const float* cls_tok= (const float*)d_in[3];
  const float* pos_emb= (const float*)d_in[4];
  const float* Wqkv   = (const float*)d_in[5];
  const float* bqkv   = (const float*)d_in[6];
  const float* Wo     = (const float*)d_in[7];
  const float* bo     = (const float*)d_in[8];
  const float* ln1_g  = (const float*)d_in[9];
  const float* ln1_b  = (const float*)d_in[10];
  const float* ln2_g  = (const float*)d_in[11];
  const float* ln2_b  = (const float*)d_in[12];
  const float* Wm1    = (const float*)d_in[13];
  const float* bm1    = (const float*)d_in[14];
  const float* Wm2    = (const float*)d_in[15];
  const float* bm2    = (const float*)d_in[16];
  const float* Wh1    = (const float*)d_in[17];
  const float* bh1    = (const float*)d_in[18];
  const float* Wh2    = (const float*)d_in[19];
  const float* bh2    = (const float*)d_in[20];
  const float* Wh3    = (const float*)d_in[21];
  const float* bh3    = (const float*)d_in[22];

  const int B = 8, D = 256, H = 16, M = 1024, L = 24, C = 1000, S = 65;
  const int Spad = 68;           // S padded to multiple of 4 (attn row stride)
  const int BS = B * S;          // 520
  const int N3 = 3 * D;          // 768
  const int HD = H * D;          // 4096
  const float scale = 1.0f / sqrtf((float)(D / H));  // (D/H)^-0.5 = 0.25

  // workspace layout (floats), ~40 MB total
  float* ws   = (float*)d_ws;
  size_t o = 0;
  float* act  = ws + o; o += (size_t)BS * D;            // residual stream
  float* zbuf = ws + o; o += (size_t)BS * D;            // LN output
  float* hbuf = ws + o; o += (size_t)BS * M;            // MLP hidden
  float* qkv  = ws + o; o += (size_t)B * H * S * N3;    // per-(b,h) qkv
  float* att  = ws + o; o += (size_t)B * H * S * Spad;  // attention probs (padded)
  float* obuf = ws + o; o += (size_t)BS * HD;           // attn out, heads concatenated
  float* h1   = ws + o; o += (size_t)B * M;
  float* h2   = ws + o; o += (size_t)B * D;

  const dim3 blk32(32);
  const int tS  = (S + 15) / 16;    // 5
  const int tBS = (BS + 15) / 16;   // 33

  { int total = BS * D;
    k_patch<<<(total + 255) / 256, 256, 0, stream>>>(x, conv_w, conv_b, cls_tok, pos_emb,
                                                     act, B, D, S); }

  for (int l = 0; l < L; ++l) {
    const float* lWqkv = Wqkv + (size_t)l * H * D * N3;
    const float* lbqkv = bqkv + (size_t)l * H * N3;
    const float* lWo   = Wo   + (size_t)l * HD * D;
    const float* lbo   = bo   + (size_t)l * D;
    const float* lWm1  = Wm1  + (size_t)l * D * M;
    const float* lbm1  = bm1  + (size_t)l * M;
    const float* lWm2  = Wm2  + (size_t)l * M * D;
    const float* lbm2  = bm2  + (size_t)l * D;

    k_ln<<<BS, 256, 0, stream>>>(act, ln1_g + (size_t)l * D, ln1_b + (size_t)l * D, zbuf, D);
    k_qkv<<<dim3(12, tS, B * H), blk32, 0, stream>>>(zbuf, lWqkv, lbqkv, qkv, H, S);
    k_scores<<<dim3(1, tS, B * H), blk32, 0, stream>>>(qkv, att, S, scale);
    { int rows = B * H * S;
      k_softmax<<<(rows + 7) / 8, 256, 0, stream>>>(att, rows, S, Spad); }
    k_av<<<dim3(4, tS, B * H), blk32, 0, stream>>>(att, qkv, obuf, H, S);
    // y1 = o @ Wo + bo + carry (in-place: each thread reads its own residual first)
    k_gemm4<4096, 256, 4096, 256, false, true>
        <<<dim3(4, tBS), blk32, 0, stream>>>(obuf, BS, lWo, lbo, act, act);
    k_ln<<<BS, 256, 0, stream>>>(act, ln2_g + (size_t)l * D, ln2_b + (size_t)l * D, zbuf, D);
    // hidden = gelu(z2 @ W1 + b1)
    k_gemm4<256, 1024, 256, 1024, true, false>
        <<<dim3(16, tBS), blk32, 0, stream>>>(zbuf, BS, lWm1, lbm1, hbuf, nullptr);
    // act = gelu(hidden @ W2 + b2) + y1
    k_gemm4<1024, 256, 1024, 256, true, true>
        <<<dim3(4, tBS), blk32, 0, stream>>>(hbuf, BS, lWm2, lbm2, act, act);
  }

  // head: cls rows are act[b*S*D + :], lda = S*D = 16640, M = B
  k_gemm4<16640, 1024, 256, 1024, true, false>
      <<<dim3(16, 1), blk32, 0, stream>>>(act, B, Wh1, bh1, h1, nullptr);
  k_gemm4<1024, 256, 1024, 256, true, false>
      <<<dim3(4, 1), blk32, 0, stream>>>(h1, B, Wh2, bh2, h2, nullptr);
  k_gemm4<256, 1000, 256, 1000, false, false>
      <<<dim3(16, 1), blk32, 0, stream>>>(h2, B, Wh3, bh3, (float*)d_out, nullptr);
}